// UnbalancedOT_24790551232773
// MI455X (gfx1250) — compile-verified
//
#include <hip/hip_runtime.h>
#include <math.h>

// Problem constants (from reference): B=4,H=128 -> 512 (b,h) slices; W=512, CC=64.
constexpr int W       = 512;
constexpr int CC      = 64;
constexpr int NBH     = 512;     // B*H
constexpr int NITER   = 10;
constexpr int STRIDE  = 68;      // 64 data floats + 4 pad floats per row (TDM pad)
constexpr float TAUc   = 0.95f;
constexpr float KAPPAc = 0.025f;            // (1-tau)/2
constexpr float XIc    = 0.05f / 1.95f;     // (1-tau)/(1+tau)
constexpr float RHOc   = 19.0f;             // eps*tau/(1-tau)
constexpr float INVRHO = 1.0f / 19.0f;
constexpr float LOG_A  = -6.2383246250395078f;  // -ln(512)
constexpr float LOG_B  = -4.1588830833596719f;  // -ln(64)

__device__ __forceinline__ float blockMax(float v, volatile float* sr) {
#pragma unroll
  for (int o = 16; o > 0; o >>= 1) v = fmaxf(v, __shfl_xor(v, o, 32));
  const int tid = threadIdx.x;
  if ((tid & 31) == 0) sr[tid >> 5] = v;
  __syncthreads();
  if (tid == 0) {
    float m = sr[0];
#pragma unroll
    for (int i = 1; i < 16; ++i) m = fmaxf(m, sr[i]);
    sr[16] = m;
  }
  __syncthreads();
  float r = sr[16];
  __syncthreads();
  return r;
}

__device__ __forceinline__ float blockSum(float v, volatile float* sr) {
#pragma unroll
  for (int o = 16; o > 0; o >>= 1) v += __shfl_xor(v, o, 32);
  const int tid = threadIdx.x;
  if ((tid & 31) == 0) sr[tid >> 5] = v;
  __syncthreads();
  if (tid == 0) {
    float s = sr[0];
#pragma unroll
    for (int i = 1; i < 16; ++i) s += sr[i];
    sr[16] = s;
  }
  __syncthreads();
  float r = sr[16];
  __syncthreads();
  return r;
}

// Max-stabilized block-wide logsumexp over participating threads' values.
__device__ __forceinline__ float blockLSE(float v, bool act, volatile float* sr) {
  const float NEG_INF = -__builtin_inff();
  float m = blockMax(act ? v : NEG_INF, sr);
  float e = act ? __expf(v - m) : 0.0f;
  float s = blockSum(e, sr);
  return m + __logf(s);
}

__global__ __launch_bounds__(512, 1)
void UnbalancedOT_24790551232773_kernel(const float* __restrict__ Cin,
                                        float* __restrict__ Out) {
  // 512*68*4 = 139264 B tile + ~4.4 KB aux; fits CDNA5's 320 KB WGP LDS.
  __shared__ float tile[W * STRIDE];
  __shared__ float fS[W];
  __shared__ float gS[CC];
  __shared__ float colM[CC];
  __shared__ float scratch[W];
  __shared__ float sred[20];

  const int tid  = threadIdx.x;
  const int wv   = tid >> 5;
  const int bh   = blockIdx.x;
  const long long base = (long long)bh * (W * CC);

  // ---- Phase 0: TDM one-shot tile load (HBM -> LDS) with hardware row padding ----
#if defined(__has_builtin) && __has_builtin(__builtin_amdgcn_tensor_load_to_lds)
  if (wv == 0) {  // uniform branch: wave 0 issues the tensor DMA (EXEC ignored by TDM)
    typedef unsigned int v4u_t __attribute__((ext_vector_type(4)));
    typedef int          v8i_t __attribute__((ext_vector_type(8)));
    typedef int          v4i_t __attribute__((ext_vector_type(4)));
    unsigned long long ga = (unsigned long long)(const void*)(Cin + base);
    unsigned int ldsOff   = (unsigned int)(unsigned long long)(void*)&tile[0];
    // D# group 0: count=1 | lds_addr | global_addr[56:0] | type=2
    v4u_t g0;
    g0.x = 1u;
    g0.y = ldsOff;
    g0.z = (unsigned int)(ga & 0xFFFFFFFFull);
    g0.w = (unsigned int)((ga >> 32) & 0x1FFFFFFull) | (2u << 30);
    // D# group 1: data_size=4B, pad_enable, pad every 64 dwords by 4 dwords,
    // tensor 64x512, tile 64x512, dim0 stride 64.
    v8i_t g1;
    g1[0] = (2 << 16) | (1 << 20) | (5 << 22) | (3 << 25);
    g1[1] = (CC & 0xFFFF) << 16;   // tensor_dim0[15:0]
    g1[2] = (W  & 0xFFFF) << 16;   // tensor_dim0[31:16]=0 | tensor_dim1[15:0]
    g1[3] = (CC & 0xFFFF) << 16;   // tensor_dim1[31:16]=0 | tile_dim0
    g1[4] = W;                     // tile_dim1 | tile_dim2=0
    g1[5] = CC;                    // tensor_dim0_stride[31:0]
    g1[6] = 0;
    g1[7] = 0;
    v4i_t gz4 = 0;                 // groups 2/3 unused (2D tensor)
    v8i_t gz8 = 0;                 // extra arg on 6-arg toolchain
    __builtin_amdgcn_tensor_load_to_lds(g0, g1, gz4, gz4, gz8, 0);
    __builtin_amdgcn_s_wait_tensorcnt(0);
  }
#else
  for (int k = tid; k < W * CC; k += 512)
    tile[(k >> 6) * STRIDE + (k & 63)] = Cin[base + k];
#endif
  __syncthreads();

  // ---- Phase 1: init potentials. f[w] = -(log_b + lse_c(-C[w,:])) ; g = 0 ----
  {
    const float* row = &tile[tid * STRIDE];
    float m = -__builtin_inff();
#pragma unroll 8
    for (int c = 0; c < CC; ++c) m = fmaxf(m, -row[c]);
    float s = 0.0f;
#pragma unroll 8
    for (int c = 0; c < CC; ++c) s += __expf(-row[c] - m);
    fS[tid] = -(LOG_B + m + __logf(s));
    if (tid < CC) gS[tid] = 0.0f;
  }
  __syncthreads();

  const int c_col  = tid & 63;       // column pass: column id
  const int chunk  = tid >> 6;       // column pass: 8 chunks of 64 rows
  const int wbase  = chunk << 6;

  for (int it = 0; it < NITER; ++it) {
    // sA = smin(old_f, log_a) = -rho*(log_a + lse_w(-f/rho))
    float sA = -RHOc * (LOG_A + blockLSE(-fS[tid] * INVRHO, true, sred));

    // ---- column LSE: L_w[c] = lse_w(f[w] - C[w,c]) (g cancels algebraically) ----
    float m = -__builtin_inff();
#pragma unroll 8
    for (int i = 0; i < 64; ++i) {
      int w = wbase + i;
      m = fmaxf(m, fS[w] - tile[w * STRIDE + c_col]);
    }
    scratch[tid] = m;
    __syncthreads();
    if (tid < CC) {
      float mm = scratch[tid];
#pragma unroll
      for (int k = 1; k < 8; ++k) mm = fmaxf(mm, scratch[tid + (k << 6)]);
      colM[tid] = mm;
    }
    __syncthreads();
    const float cm = colM[c_col];
    float s = 0.0f;
#pragma unroll 8
    for (int i = 0; i < 64; ++i) {
      int w = wbase + i;
      s += __expf(fS[w] - tile[w * STRIDE + c_col] - cm);
    }
    scratch[tid] = s;
    __syncthreads();
    if (tid < CC) {
      float ss = scratch[tid];
#pragma unroll
      for (int k = 1; k < 8; ++k) ss += scratch[tid + (k << 6)];
      float Lw = colM[tid] + __logf(ss);
      gS[tid] = TAUc * (LOG_B - Lw) - KAPPAc * sA;   // new_g pre-XI
    }
    __syncthreads();

    // sB = smin(new_g_pre, log_b); new_g += xi*sB
    float sB = -RHOc * (LOG_B + blockLSE(tid < CC ? -gS[tid] * INVRHO : 0.0f,
                                         tid < CC, sred));
    if (tid < CC) gS[tid] += XIc * sB;
    __syncthreads();

    // ---- row LSE: L_c[w] = lse_c(g[c] - C[w,:]) (f cancels algebraically) ----
    const float* row = &tile[tid * STRIDE];
    float m2 = -__builtin_inff();
#pragma unroll 8
    for (int c = 0; c < CC; ++c) m2 = fmaxf(m2, gS[c] - row[c]);
    float s2 = 0.0f;
#pragma unroll 8
    for (int c = 0; c < CC; ++c) s2 += __expf(gS[c] - row[c] - m2);
    float nf = TAUc * (LOG_A - (m2 + __logf(s2)));

    // sC = smin(new_g_final, log_b)
    float sC = -RHOc * (LOG_B + blockLSE(tid < CC ? -gS[tid] * INVRHO : 0.0f,
                                         tid < CC, sred));
    nf -= KAPPAc * sC;
    // sD = smin(new_f, log_a)
    float sD = -RHOc * (LOG_A + blockLSE(-nf * INVRHO, true, sred));
    nf += XIc * sD;
    fS[tid] = nf;
    __syncthreads();
  }

  // ---- Final: out[w,c] = exp(f[w] + g[c] - C[w,c]) ; fully coalesced stores ----
#pragma unroll 4
  for (int k = 0; k < 64; ++k) {
    int l = tid + (k << 9);
    int w = l >> 6, c = l & 63;
    Out[base + l] = __expf(fS[w] + gS[c] - tile[w * STRIDE + c]);
  }
}

extern "C" void kernel_launch(void* const* d_in, const int* in_sizes, int n_in,
                              void* d_out, int out_size, void* d_ws, size_t ws_size,
                              hipStream_t stream) {
  const float* C = (const float*)d_in[0];
  float* out = (float*)d_out;
  hipLaunchKernelGGL(UnbalancedOT_24790551232773_kernel,
                     dim3(NBH), dim3(512), 0, stream, C, out);
}